// AdvancedTransformerBlock_73375221284959
// MI455X (gfx1250) — compile-verified
//
#include <hip/hip_runtime.h>
#include <math.h>

typedef __attribute__((ext_vector_type(2))) float v2f;
typedef __attribute__((ext_vector_type(8))) float v8f;

#define BM 128
#define BN 128
#define BK 16
#define LDP 18   // padded K-stride (even -> 8B-aligned v2f loads, conflict-free)
#define QH 12

// ---------------------------------------------------------------------------
// Generic batched WMMA GEMM:  C[z] = epilogue( alpha * A[z] . op(B[z]) + bias )
//   A: M x K row-major (lda), op(B): K x N  (transB: B stored N x K row-major)
//   z decomposed as (zb = z / nH, zh = z % nH) with independent strides.
//   accum==1:  C += rowscale[r*rsStride] * val   (MoE gate-weighted accumulate)
//   accum==0:  C  = val (+ resid[r*ldc+c] if resid != nullptr)
//   act==1: exact GELU
// Tiles: 128x128 block tile, BK=16, 8 waves (4x2), each wave 32x64
// (2x4 tiles of 16x16), inner K step 4 via V_WMMA_F32_16X16X4_F32.
// Both LDS tiles stored K-fastest with stride LDP so each fragment is one
// aligned ds_load_b64.
// ---------------------------------------------------------------------------
__global__ __launch_bounds__(256) void gemm_wmma_kernel(
    const float* __restrict__ A, long lda, long sAb, long sAh,
    const float* __restrict__ B, long ldb, long sBb, long sBh, int transB,
    const float* __restrict__ bias,
    const float* __restrict__ resid,
    const float* __restrict__ rowscale, int rsStride,
    float* __restrict__ C, long ldc, long sCb, long sCh,
    int M, int N, int K, int nH, float alpha, int act, int accum)
{
  __shared__ float As[BM][LDP];   // As[row][k]
  __shared__ float Bs[BN][LDP];   // Bs[col][k]

  const int zb = blockIdx.z / nH;
  const int zh = blockIdx.z % nH;
  A += (long)zb * sAb + (long)zh * sAh;
  B += (long)zb * sBb + (long)zh * sBh;
  C += (long)zb * sCb + (long)zh * sCh;

  const int tid   = threadIdx.x;
  const int lane  = tid & 31;
  const int wave  = tid >> 5;
  const int wm    = wave >> 1;    // 0..3 -> 32-row group
  const int wn    = wave & 1;     // 0..1 -> 64-col group
  const int lrow  = lane & 15;
  const int lhalf = lane >> 4;    // 0: K pair {0,1}, 1: K pair {2,3}

  const int rowBase = blockIdx.y * BM;
  const int colBase = blockIdx.x * BN;

  v8f acc[2][4];
#pragma unroll
  for (int i = 0; i < 2; ++i)
#pragma unroll
    for (int j = 0; j < 4; ++j)
#pragma unroll
      for (int e = 0; e < 8; ++e) acc[i][j][e] = 0.0f;

  for (int k0 = 0; k0 < K; k0 += BK) {
    // stage A tile (BM x BK): K-fastest, 16-float contiguous reads per row
    for (int e = tid; e < BM * BK; e += 256) {
      int r = e / BK, c = e % BK;
      int gr = rowBase + r;
      As[r][c] = (gr < M) ? A[(long)gr * lda + (k0 + c)] : 0.0f;
    }
    // stage B tile as Bs[col][k]
    if (transB) {
      for (int e = tid; e < BN * BK; e += 256) {
        int n = e / BK, kk = e % BK;
        int gn = colBase + n;
        Bs[n][kk] = (gn < N) ? B[(long)gn * ldb + (k0 + kk)] : 0.0f;
      }
    } else {
      for (int e = tid; e < BN * BK; e += 256) {
        int kk = e / BN, n = e % BN;
        int gn = colBase + n;
        Bs[n][kk] = (gn < N) ? B[(long)(k0 + kk) * ldb + gn] : 0.0f;
      }
    }
    // prefetch next K tile into cache while this one is consumed
    int nk = k0 + BK;
    if (nk < K) {
      if (tid < BM) {
        int gr = rowBase + tid;
        if (gr < M) __builtin_prefetch((const void*)&A[(long)gr * lda + nk], 0, 1);
      } else if (transB) {
        int gn = colBase + (tid - BM);
        if (gn < N) __builtin_prefetch((const void*)&B[(long)gn * ldb + nk], 0, 1);
      } else {
        int kk = tid - BM;
        if (kk < BK) __builtin_prefetch((const void*)&B[(long)(nk + kk) * ldb + colBase], 0, 1);
      }
    }
    __syncthreads();

#pragma unroll
    for (int kk = 0; kk < BK; kk += 4) {
      const int kc = kk + 2 * lhalf;          // even -> 8B aligned
      v2f af[2], bf[4];
#pragma unroll
      for (int i = 0; i < 2; ++i) {
        int ar = wm * 32 + i * 16 + lrow;     // A: lane = M row
        af[i] = *(const v2f*)&As[ar][kc];     // one ds_load_b64
      }
#pragma unroll
      for (int j = 0; j < 4; ++j) {
        int bc = wn * 64 + j * 16 + lrow;     // B: lane = N col
        bf[j] = *(const v2f*)&Bs[bc][kc];     // one ds_load_b64
      }
#pragma unroll
      for (int i = 0; i < 2; ++i)
#pragma unroll
        for (int j = 0; j < 4; ++j)
          acc[i][j] = __builtin_amdgcn_wmma_f32_16x16x4_f32(
              false, af[i], false, bf[j], (short)0, acc[i][j], false, false);
    }
    __syncthreads();
  }

  // epilogue; C/D layout: VGPR v -> row v + 8*lhalf, col = lane&15
#pragma unroll
  for (int i = 0; i < 2; ++i)
#pragma unroll
    for (int j = 0; j < 4; ++j) {
      int cg = colBase + wn * 64 + j * 16 + lrow;
      if (cg >= N) continue;
      float bv = bias ? bias[cg] : 0.0f;
#pragma unroll
      for (int v = 0; v < 8; ++v) {
        int rg = rowBase + wm * 32 + i * 16 + v + lhalf * 8;
        if (rg >= M) continue;
        float val = alpha * acc[i][j][v] + bv;
        if (act == 1)  // exact GELU
          val = 0.5f * val * (1.0f + erff(val * 0.70710678118654752f));
        long idx = (long)rg * ldc + cg;
        if (accum == 1) {
          C[idx] += rowscale[(long)rg * rsStride] * val;
        } else {
          if (resid) val += resid[idx];
          C[idx] = val;
        }
      }
    }
}

// ---------------------------------------------------------------------------
// PHM weight materialization: W[a*So+c, b*Si+d] = sum_i A4[i,a,b] * S[i,c,d]
// ---------------------------------------------------------------------------
__global__ void phm_kernel(float* __restrict__ W, const float* __restrict__ A4,
                           const float* __restrict__ S, int So, int Si,
                           int outF, int inF)
{
  long idx = (long)blockIdx.x * blockDim.x + threadIdx.x;
  long total = (long)outF * inF;
  if (idx >= total) return;
  int o = (int)(idx / inF), j = (int)(idx % inF);
  int a = o / So, c = o % So, b = j / Si, d = j % Si;
  float s = 0.0f;
#pragma unroll
  for (int i = 0; i < 4; ++i)
    s += A4[i * 16 + a * 4 + b] * S[((long)i * So + c) * Si + d];
  W[idx] = s;
}

// ---------------------------------------------------------------------------
// LayerNorm (biased variance, eps=1e-5), one 256-thread block per row
// ---------------------------------------------------------------------------
__global__ __launch_bounds__(256) void ln_kernel(const float* __restrict__ x,
    const float* __restrict__ g, const float* __restrict__ b,
    float* __restrict__ out, int C)
{
  __shared__ float sh[256];
  long row = blockIdx.x;
  const float* xr = x + row * (long)C;
  float s = 0.0f, s2 = 0.0f;
  for (int i = threadIdx.x; i < C; i += 256) { float v = xr[i]; s += v; s2 += v * v; }
  sh[threadIdx.x] = s; __syncthreads();
  for (int st = 128; st > 0; st >>= 1) {
    if (threadIdx.x < st) sh[threadIdx.x] += sh[threadIdx.x + st];
    __syncthreads();
  }
  float S = sh[0]; __syncthreads();
  sh[threadIdx.x] = s2; __syncthreads();
  for (int st = 128; st > 0; st >>= 1) {
    if (threadIdx.x < st) sh[threadIdx.x] += sh[threadIdx.x + st];
    __syncthreads();
  }
  float S2 = sh[0];
  float mean = S / C;
  float var  = S2 / C - mean * mean;
  float inv  = rsqrtf(var + 1e-5f);
  for (int i = threadIdx.x; i < C; i += 256)
    out[row * (long)C + i] = (xr[i] - mean) * inv * g[i] + b[i];
}

// ---------------------------------------------------------------------------
// Head mixing over 12 quantum heads on (b,n,m) planes.
// mode 0: out[i] = sum_j w[i,j] in[j]                 (superposition)
// mode 1: out[i] = tanh(in[i] + sum_j w[i,j] in[j])   (entanglement)
// ---------------------------------------------------------------------------
__global__ __launch_bounds__(256) void headmix_kernel(const float* __restrict__ in,
    float* __restrict__ out, const float* __restrict__ w,
    long plane, int mode, long total)
{
  __shared__ float ws[QH * QH];
  if (threadIdx.x < QH * QH) ws[threadIdx.x] = w[threadIdx.x];
  __syncthreads();
  long idx = (long)blockIdx.x * blockDim.x + threadIdx.x;
  if (idx >= total) return;
  long bq = idx / plane;
  long p  = idx % plane;
  const float* ib = in + bq * QH * plane + p;
  float* ob       = out + bq * QH * plane + p;
  float vals[QH];
#pragma unroll
  for (int j = 0; j < QH; ++j) vals[j] = ib[(long)j * plane];
#pragma unroll
  for (int i = 0; i < QH; ++i) {
    float a = 0.0f;
#pragma unroll
    for (int j = 0; j < QH; ++j) a += ws[i * QH + j] * vals[j];
    ob[(long)i * plane] = mode ? tanhf(vals[i] + a) : a;
  }
}

// row softmax in place, one block per row
__global__ __launch_bounds__(256) void softmax_kernel(float* __restrict__ data, int W)
{
  __shared__ float sh[256];
  long row = blockIdx.x;
  float* p = data + row * (long)W;
  float m = -3.4e38f;
  for (int i = threadIdx.x; i < W; i += 256) m = fmaxf(m, p[i]);
  sh[threadIdx.x] = m; __syncthreads();
  for (int st = 128; st > 0; st >>= 1) {
    if (threadIdx.x < st) sh[threadIdx.x] = fmaxf(sh[threadIdx.x], sh[threadIdx.x + st]);
    __syncthreads();
  }
  float mx = sh[0]; __syncthreads();
  float s = 0.0f;
  for (int i = threadIdx.x; i < W; i += 256) { float e = expf(p[i] - mx); p[i] = e; s += e; }
  sh[threadIdx.x] = s; __syncthreads();
  for (int st = 128; st > 0; st >>= 1) {
    if (threadIdx.x < st) sh[threadIdx.x] += sh[threadIdx.x + st];
    __syncthreads();
  }
  float inv = 1.0f / sh[0];
  for (int i = threadIdx.x; i < W; i += 256) p[i] *= inv;
}

// top-2 router gates (matches jax.lax.top_k tie-breaking: first occurrence)
__global__ void gates_kernel(const float* __restrict__ logits,
    const float* __restrict__ dr, const int* __restrict__ did,
    float* __restrict__ gates, int T)
{
  int t = blockIdx.x * blockDim.x + threadIdx.x;
  if (t >= T) return;
  int d = did[0];
  float l[8];
#pragma unroll
  for (int e = 0; e < 8; ++e) l[e] = logits[t * 8 + e] + dr[d * 8 + e];
  int i0 = 0; float v0 = l[0];
#pragma unroll
  for (int e = 1; e < 8; ++e) if (l[e] > v0) { v0 = l[e]; i0 = e; }
  int i1 = -1; float v1 = -3.4e38f;
#pragma unroll
  for (int e = 0; e < 8; ++e) if (e != i0 && l[e] > v1) { v1 = l[e]; i1 = e; }
  float e1 = expf(v1 - v0);
  float den = 1.0f + e1;
#pragma unroll
  for (int e = 0; e < 8; ++e) gates[t * 8 + e] = 0.0f;
  gates[t * 8 + i0] = 1.0f / den;
  gates[t * 8 + i1] = e1 / den;
}

__global__ void ew_add_kernel(float* __restrict__ out, const float* __restrict__ a,
                              const float* __restrict__ b, long n)
{
  long i = (long)blockIdx.x * blockDim.x + threadIdx.x;
  if (i < n) out[i] = a[i] + (b ? b[i] : 0.0f);
}
__global__ void ew_acc_kernel(float* __restrict__ out, const float* __restrict__ a, long n)
{
  long i = (long)blockIdx.x * blockDim.x + threadIdx.x;
  if (i < n) out[i] += a[i];
}
__global__ void ew_zero_kernel(float* __restrict__ p, long n)
{
  long i = (long)blockIdx.x * blockDim.x + threadIdx.x;
  if (i < n) p[i] = 0.0f;
}

// ---------------------------------------------------------------------------
// host-side helpers
// ---------------------------------------------------------------------------
static inline void launch_gemm(hipStream_t s,
    const float* A, long lda, long sAb, long sAh,
    const float* B, long ldb, long sBb, long sBh, int transB,
    const float* bias, const float* resid,
    const float* rowscale, int rsStride,
    float* C, long ldc, long sCb, long sCh,
    int M, int N, int K, int nz, int nH, float alpha, int act, int accum)
{
  dim3 g((N + BN - 1) / BN, (M + BM - 1) / BM, nz);
  gemm_wmma_kernel<<<g, 256, 0, s>>>(A, lda, sAb, sAh, B, ldb, sBb, sBh, transB,
                                     bias, resid, rowscale, rsStride,
                                     C, ldc, sCb, sCh, M, N, K, nH, alpha, act, accum);
}
static inline void launch_phm(hipStream_t s, float* W, const float* A4,
                              const float* S, int So, int Si, int outF, int inF)
{
  long total = (long)outF * inF;
  phm_kernel<<<dim3((unsigned)((total + 255) / 256)), 256, 0, s>>>(W, A4, S, So, Si, outF, inF);
}

extern "C" void kernel_launch(void* const* d_in, const int* in_sizes, int n_in,
                              void* d_out, int out_size, void* d_ws, size_t ws_size,
                              hipStream_t stream)
{
  (void)in_sizes; (void)n_in; (void)out_size; (void)ws_size;
  const float* x         = (const float*)d_in[0];
  const float* ln1_g     = (const float*)d_in[1];
  const float* ln1_b     = (const float*)d_in[2];
  const float* ln2_g     = (const float*)d_in[3];
  const float* ln2_b     = (const float*)d_in[4];
  const float* ln3_g     = (const float*)d_in[5];
  const float* ln3_b     = (const float*)d_in[6];
  const float* qkv_A     = (const float*)d_in[7];
  const float* qkv_S     = (const float*)d_in[8];
  const float* qkv_b     = (const float*)d_in[9];
  const float* proj_A    = (const float*)d_in[10];
  const float* proj_S    = (const float*)d_in[11];
  const float* proj_b    = (const float*)d_in[12];
  const float* w_sup     = (const float*)d_in[13];
  const float* w_ent     = (const float*)d_in[14];
  const float* router_A  = (const float*)d_in[15];
  const float* router_S  = (const float*)d_in[16];
  const float* router_b  = (const float*)d_in[17];
  const float* dom_rt    = (const float*)d_in[18];
  const float* exp_A     = (const float*)d_in[19];
  const float* exp_S     = (const float*)d_in[20];
  const float* exp_b     = (const float*)d_in[21];
  const float* exp_nd_w  = (const float*)d_in[22];
  const float* exp_nd_b  = (const float*)d_in[23];
  const float* mems[3]   = { (const float*)d_in[24], (const float*)d_in[25], (const float*)d_in[26] };
  const float* mha_in_w  = (const float*)d_in[27];
  const float* mha_in_b  = (const float*)d_in[28];
  const float* mha_out_w = (const float*)d_in[29];
  const float* mha_out_b = (const float*)d_in[30];
  const float* proc_w    = (const float*)d_in[31];
  const float* proc_b    = (const float*)d_in[32];
  const int*   dom_id    = (const int*)d_in[33];
  float* out = (float*)d_out;

  const int B = 4, N = 512, C = 768, T = B * N;          // 2048 tokens
  const long NM = (long)N * N;                           // 262144 per-head plane

  float* ws = (float*)d_ws;
  size_t off = 0;
  auto alloc = [&](size_t n) { float* p = ws + off; off += (n + 63) & ~(size_t)63; return p; };
  float* bufLn    = alloc((size_t)T * C);        // LN outputs / fractal tmp
  float* bufBig   = alloc((size_t)T * 3072);     // qkv (T x 2304), later expert h (T x 3072)
  float* bufAttn  = alloc((size_t)B * QH * NM);  // attn scores / ent / probs, later mem-attn
  float* bufSup   = alloc((size_t)B * QH * NM);  // superposition buffer
  float* bufO     = alloc((size_t)T * C);        // attention context
  float* Wqkv     = alloc((size_t)2304 * 768);
  float* Wproj    = alloc((size_t)768 * 768);
  float* attended = alloc((size_t)T * C);
  float* Wrouter  = alloc((size_t)8 * 768);
  float* logits   = alloc((size_t)T * 8);
  float* gates    = alloc((size_t)T * 8);
  float* Wexp     = alloc((size_t)3072 * 768);
  float* moeacc   = alloc((size_t)T * C);
  float* eo       = alloc((size_t)T * C);
  float* qbuf     = alloc((size_t)T * C);
  float* kproj    = alloc((size_t)256 * 768);
  float* vproj    = alloc((size_t)256 * 768);
  float* acur     = alloc((size_t)T * C);
  float* mo       = alloc((size_t)T * C);

  const long TC = (long)T * C;
  dim3 ewg((unsigned)((TC + 255) / 256));

  // ---- quantum attention ----
  ln_kernel<<<T, 256, 0, stream>>>(x, ln1_g, ln1_b, bufLn, C);
  launch_phm(stream, Wqkv, qkv_A, qkv_S, 576, 192, 2304, 768);
  launch_gemm(stream, bufLn, 768, 0, 0, Wqkv, 768, 0, 0, 1, qkv_b, nullptr, nullptr, 0,
              bufBig, 2304, 0, 0, T, 2304, 768, 1, 1, 1.0f, 0, 0);
  // scores[b,h] = q . k^T * 1/8   (z = b*12+h)
  launch_gemm(stream, bufBig, 2304, (long)N * 2304, 64,
              bufBig + 768, 2304, (long)N * 2304, 64, 1,
              nullptr, nullptr, nullptr, 0,
              bufAttn, N, QH * NM, NM, N, N, 64, B * QH, QH, 0.125f, 0, 0);
  {
    long total = (long)B * NM;
    dim3 g((unsigned)((total + 255) / 256));
    headmix_kernel<<<g, 256, 0, stream>>>(bufAttn, bufSup, w_sup, NM, 0, total);
    headmix_kernel<<<g, 256, 0, stream>>>(bufSup, bufAttn, w_ent, NM, 1, total);
  }
  softmax_kernel<<<B * QH * N, 256, 0, stream>>>(bufAttn, N);
  // o[b,n,h,:] = probs . v
  launch_gemm(stream, bufAttn, N, QH * NM, NM,
              bufBig + 1536, 2304, (long)N * 2304, 64, 0,
              nullptr, nullptr, nullptr, 0,
              bufO, 768, (long)N * 768, 64, N, 64, N, B * QH, QH, 1.0f, 0, 0);
  launch_phm(stream, Wproj, proj_A, proj_S, 192, 192, 768, 768);
  launch_gemm(stream, bufO, 768, 0, 0, Wproj, 768, 0, 0, 1, proj_b, x, nullptr, 0,
              attended, 768, 0, 0, T, 768, 768, 1, 1, 1.0f, 0, 0);

  // ---- MoE ----
  ln_kernel<<<T, 256, 0, stream>>>(attended, ln2_g, ln2_b, bufLn, C);
  launch_phm(stream, Wrouter, router_A, router_S, 2, 192, 8, 768);
  launch_gemm(stream, bufLn, 768, 0, 0, Wrouter, 768, 0, 0, 1, router_b, nullptr, nullptr, 0,
              logits, 8, 0, 0, T, 8, 768, 1, 1, 1.0f, 0, 0);
  gates_kernel<<<(T + 255) / 256, 256, 0, stream>>>(logits, dom_rt, dom_id, gates, T);
  ew_zero_kernel<<<ewg, 256, 0, stream>>>(moeacc, TC);
  for (int e = 0; e < 8; ++e) {
    launch_phm(stream, Wexp, exp_A + (long)e * 64, exp_S + (long)e * 4 * 768 * 192,
               768, 192, 3072, 768);
    launch_gemm(stream, bufLn, 768, 0, 0, Wexp, 768, 0, 0, 1, exp_b + (long)e * 3072,
                nullptr, nullptr, 0, bufBig, 3072, 0, 0, T, 3072, 768, 1, 1, 1.0f, 1, 0);
    launch_gemm(stream, bufBig, 3072, 0, 0, exp_nd_w + (long)e * 768 * 3072, 3072, 0, 0, 1,
                exp_nd_b + (long)e * 768, nullptr, gates + e, 8,
                moeacc, 768, 0, 0, T, 768, 3072, 1, 1, 1.0f, 0, 1);
  }
  ew_add_kernel<<<ewg, 256, 0, stream>>>(eo, attended, moeacc, TC);
  ew_add_kernel<<<ewg, 256, 0, stream>>>(mo, eo, nullptr, TC);   // mo = eo

  // ---- fractal memory (3 levels of standard MHA over memory banks) ----
  const int Ml[3] = { 64, 128, 256 };
  const float* xin = eo;
  const float ascale = 1.0f / sqrtf(96.0f);
  for (int l = 0; l < 3; ++l) {
    const float* iw = mha_in_w + (long)l * 2304 * 768;
    const float* ib = mha_in_b + (long)l * 2304;
    int M_mem = Ml[l];
    // q = xin @ iw_q^T + ib_q ; k,v from memory bank
    launch_gemm(stream, xin, 768, 0, 0, iw, 768, 0, 0, 1, ib, nullptr, nullptr, 0,
                qbuf, 768, 0, 0, T, 768, 768, 1, 1, 1.0f, 0, 0);
    launch_gemm(stream, mems[l], 768, 0, 0, iw + (long)768 * 768, 768, 0, 0, 1, ib + 768,
                nullptr, nullptr, 0, kproj, 768, 0, 0, M_mem, 768, 768, 1, 1, 1.0f, 0, 0);
    launch_gemm(stream, mems[l], 768, 0, 0, iw + (long)1536 * 768, 768, 0, 0, 1, ib + 1536,
                nullptr, nullptr, 0, vproj, 768, 0, 0, M_mem, 768, 768, 1, 1, 1.0f, 0, 0);
    // scores[b,h] (8 heads of dim 96), softmax over memory dim, context
    launch_gemm(stream, qbuf, 768, (long)N * 768, 96,
                kproj, 768, 0, 96, 1, nullptr, nullptr, nullptr, 0,
                bufAttn, M_mem, (long)8 * N * M_mem, (long)N * M_mem,
                N, M_mem, 96, B * 8, 8, ascale, 0, 0);
    softmax_kernel<<<B * 8 * N, 256, 0, stream>>>(bufAttn, M_mem);
    launch_gemm(stream, bufAttn, M_mem, (long)8 * N * M_mem, (long)N * M_mem,
                vproj, 768, 0, 96, 0, nullptr, nullptr, nullptr, 0,
                bufO, 768, (long)N * 768, 96, N, 96, M_mem, B * 8, 8, 1.0f, 0, 0);
    // out proj then processor
    launch_gemm(stream, bufO, 768, 0, 0, mha_out_w + (long)l * 768 * 768, 768, 0, 0, 1,
                mha_out_b + (long)l * 768, nullptr, nullptr, 0,
                bufLn, 768, 0, 0, T, 768, 768, 1, 1, 1.0f, 0, 0);
    launch_gemm(stream, bufLn, 768, 0, 0, proc_w + (long)l * 768 * 768, 768, 0, 0, 1,
                proc_b + (long)l * 768, nullptr, nullptr, 0,
                acur, 768, 0, 0, T, 768, 768, 1, 1, 1.0f, 0, 0);
    ew_acc_kernel<<<ewg, 256, 0, stream>>>(mo, acur, TC);
    xin = acur;
  }

  // ---- final layernorm ----
  ln_kernel<<<T, 256, 0, stream>>>(mo, ln3_g, ln3_b, out, C);
}